// GraphAttention2_64261300682765
// MI455X (gfx1250) — compile-verified
//
#include <hip/hip_runtime.h>

// GAT layer for MI455X (gfx1250, wave32).
// ws layout (floats): ft[N*128] | a1[N*8] | a2[N*8] | esum[N*8] | emax_enc[N*8] (~32 MB)

#define IN_DIM 128
#define ALPHA 0.2f

typedef __attribute__((ext_vector_type(2))) float v2f;
typedef __attribute__((ext_vector_type(8))) float v8f;

// ---------- helpers ----------
__device__ __forceinline__ float lrelu(float v) { return v > 0.f ? v : ALPHA * v; }

// order-preserving float -> uint encoding for atomicMax-based segment max
__device__ __forceinline__ unsigned enc_f32(float f) {
  unsigned u = __float_as_uint(f);
  return (u & 0x80000000u) ? ~u : (u | 0x80000000u);
}
__device__ __forceinline__ float dec_f32(unsigned u) {
  return (u & 0x80000000u) ? __uint_as_float(u & 0x7FFFFFFFu) : __uint_as_float(~u);
}

__device__ __forceinline__ float edge_logit(const float* __restrict__ a1,
                                            const float* __restrict__ a2,
                                            long long s, long long d, int h) {
  return lrelu(a1[s * 8 + h] + a2[d * 8 + h]);
}

// ---------- kernel 1: fused fc1 -> leaky_relu -> fc2 via V_WMMA_F32_16X16X4_F32 ----------
// block = 128 threads = 4 waves; each wave owns one 16-node row tile.
__global__ __launch_bounds__(128) void gat_gemm_wmma(const float* __restrict__ x,
                                                     const float* __restrict__ W1,
                                                     const float* __restrict__ W2,
                                                     float* __restrict__ ft, int N) {
  __shared__ float ldsX[4][16 * IN_DIM];  // staged x tile per wave
  __shared__ float ldsH[4][16 * IN_DIM];  // h2 = lrelu(x@W1^T) tile per wave

  const int lane = threadIdx.x & 31;
  const int wave = threadIdx.x >> 5;
  const int rowBase = (blockIdx.x * 4 + wave) * 16;
  float* lx = ldsX[wave];
  float* lh = ldsH[wave];

  // stage 16x128 x tile into LDS (coalesced float4), clamp OOB rows
  for (int r = 0; r < 16; ++r) {
    int gr = rowBase + r;
    int src = (gr < N) ? gr : 0;
    float4 v = ((const float4*)(x + (size_t)src * IN_DIM))[lane];
    ((float4*)(lx + r * IN_DIM))[lane] = v;
  }
  __syncthreads();

  // WMMA fragment coordinates (ISA 7.12.2): A 16x4 f32, C/D 16x16 f32
  const int m = lane & 15;            // A row / B,C column index
  const int kb = (lane >> 4) << 1;    // k sub-offset: 0 (lanes 0-15) or 2 (lanes 16-31)
  const int mrow = (lane >> 4) * 8;   // C/D row offset for this half-wave

  // GEMM1: h2 = lrelu(x @ W1^T); 8 column tiles of 16
  for (int nt = 0; nt < 8; ++nt) {
    v8f c = {0.f, 0.f, 0.f, 0.f, 0.f, 0.f, 0.f, 0.f};
    const float* w1row = W1 + (size_t)(nt * 16 + m) * IN_DIM;  // B[k][n] = W1[n][k]
#pragma unroll 8
    for (int k0 = 0; k0 < IN_DIM; k0 += 4) {
      float2 af = *(const float2*)(lx + m * IN_DIM + k0 + kb);
      float2 bf = *(const float2*)(w1row + k0 + kb);
      v2f a; a.x = af.x; a.y = af.y;
      v2f b; b.x = bf.x; b.y = bf.y;
      c = __builtin_amdgcn_wmma_f32_16x16x4_f32(false, a, false, b, (short)0, c,
                                                false, false);
    }
#pragma unroll
    for (int r = 0; r < 8; ++r)
      lh[(mrow + r) * IN_DIM + nt * 16 + m] = lrelu(c[r]);
  }
  __syncthreads();

  // GEMM2: ft = h2 @ W2^T
  for (int nt = 0; nt < 8; ++nt) {
    v8f c = {0.f, 0.f, 0.f, 0.f, 0.f, 0.f, 0.f, 0.f};
    const float* w2row = W2 + (size_t)(nt * 16 + m) * IN_DIM;
#pragma unroll 8
    for (int k0 = 0; k0 < IN_DIM; k0 += 4) {
      float2 af = *(const float2*)(lh + m * IN_DIM + k0 + kb);
      float2 bf = *(const float2*)(w2row + k0 + kb);
      v2f a; a.x = af.x; a.y = af.y;
      v2f b; b.x = bf.x; b.y = bf.y;
      c = __builtin_amdgcn_wmma_f32_16x16x4_f32(false, a, false, b, (short)0, c,
                                                false, false);
    }
#pragma unroll
    for (int r = 0; r < 8; ++r) {
      int grow = rowBase + mrow + r;
      if (grow < N) ft[(size_t)grow * IN_DIM + nt * 16 + m] = c[r];
    }
  }
}

// ---------- kernel 2: per-node attention halves a1, a2 ----------
__global__ void gat_node_attn(const float* __restrict__ ft,
                              const float* __restrict__ attn_l,
                              const float* __restrict__ attn_r,
                              float* __restrict__ a1, float* __restrict__ a2, int N) {
  int i = blockIdx.x * blockDim.x + threadIdx.x;  // over N*8 (node, head)
  if (i >= N * 8) return;
  int h = i & 7, n = i >> 3;
  const float4* f = (const float4*)(ft + (size_t)n * IN_DIM + h * 16);
  const float4* al = (const float4*)(attn_l + h * 16);
  const float4* ar = (const float4*)(attn_r + h * 16);
  float s1 = 0.f, s2 = 0.f;
#pragma unroll
  for (int j = 0; j < 4; ++j) {
    float4 fv = f[j], lv = al[j], rv = ar[j];
    s1 += fv.x * lv.x + fv.y * lv.y + fv.z * lv.z + fv.w * lv.w;
    s2 += fv.x * rv.x + fv.y * rv.y + fv.z * rv.z + fv.w * rv.w;
  }
  a1[i] = s1;
  a2[i] = s2;
}

// ---------- kernel 3: zero accumulators ----------
__global__ void gat_init(float* __restrict__ out, float* __restrict__ esum,
                         unsigned* __restrict__ emax, int N) {
  int i = blockIdx.x * blockDim.x + threadIdx.x;
  if (i < N * IN_DIM) out[i] = 0.f;
  if (i < N * 8) { esum[i] = 0.f; emax[i] = 0u; }  // 0 < enc(-inf): safe floor
}

// ---------- kernel 4: segment max over incoming edges ----------
__global__ void gat_edge_max(const long long* __restrict__ ei,
                             const float* __restrict__ a1, const float* __restrict__ a2,
                             unsigned* __restrict__ emax, long long E) {
  long long i = blockIdx.x * (long long)blockDim.x + threadIdx.x;
  if (i >= E * 8) return;
  int h = (int)(i & 7);
  long long e = i >> 3;
  long long s = ei[e], d = ei[E + e];
  atomicMax(&emax[d * 8 + h], enc_f32(edge_logit(a1, a2, s, d, h)));
}

// ---------- kernel 5: segment sum of exp(e - max) ----------
__global__ void gat_edge_expsum(const long long* __restrict__ ei,
                                const float* __restrict__ a1, const float* __restrict__ a2,
                                const unsigned* __restrict__ emax,
                                float* __restrict__ esum, long long E) {
  long long i = blockIdx.x * (long long)blockDim.x + threadIdx.x;
  if (i >= E * 8) return;
  int h = (int)(i & 7);
  long long e = i >> 3;
  long long s = ei[e], d = ei[E + e];
  float lg = edge_logit(a1, a2, s, d, h);
  atomicAdd(&esum[d * 8 + h], expf(lg - dec_f32(emax[d * 8 + h])));
}

// ---------- kernel 6: weighted scatter-sum aggregation ----------
__global__ void gat_edge_aggregate(const long long* __restrict__ ei,
                                   const float* __restrict__ a1, const float* __restrict__ a2,
                                   const unsigned* __restrict__ emax,
                                   const float* __restrict__ esum,
                                   const float* __restrict__ ft,
                                   float* __restrict__ out, long long E) {
  long long i = blockIdx.x * (long long)blockDim.x + threadIdx.x;
  if (i >= E * 8) return;
  int h = (int)(i & 7);
  long long e = i >> 3;
  long long s = ei[e], d = ei[E + e];
  float lg = edge_logit(a1, a2, s, d, h);
  float a = expf(lg - dec_f32(emax[d * 8 + h])) / esum[d * 8 + h];
  const float4* fv = (const float4*)(ft + (size_t)s * IN_DIM + h * 16);
  float* o = out + (size_t)d * IN_DIM + h * 16;
#pragma unroll
  for (int j = 0; j < 4; ++j) {
    float4 v = fv[j];
    atomicAdd(o + j * 4 + 0, a * v.x);
    atomicAdd(o + j * 4 + 1, a * v.y);
    atomicAdd(o + j * 4 + 2, a * v.z);
    atomicAdd(o + j * 4 + 3, a * v.w);
  }
}

extern "C" void kernel_launch(void* const* d_in, const int* in_sizes, int n_in,
                              void* d_out, int out_size, void* d_ws, size_t ws_size,
                              hipStream_t stream) {
  const float* x = (const float*)d_in[0];
  const float* W1 = (const float*)d_in[1];
  const float* W2 = (const float*)d_in[2];
  const float* attn_l = (const float*)d_in[3];
  const float* attn_r = (const float*)d_in[4];
  const long long* ei = (const long long*)d_in[5];  // int64 edge_index [2, E]

  int N = in_sizes[0] / IN_DIM;
  long long E = in_sizes[5] / 2;
  float* out = (float*)d_out;

  float* ft = (float*)d_ws;                       // N*128
  float* a1 = ft + (size_t)N * IN_DIM;            // N*8
  float* a2 = a1 + (size_t)N * 8;                 // N*8
  float* esum = a2 + (size_t)N * 8;               // N*8
  unsigned* emax = (unsigned*)(esum + (size_t)N * 8);  // N*8

  int gemmBlocks = (N + 63) / 64;  // 4 waves/block * 16 rows/wave
  gat_gemm_wmma<<<gemmBlocks, 128, 0, stream>>>(x, W1, W2, ft, N);

  int initBlocks = (N * IN_DIM + 255) / 256;
  gat_init<<<initBlocks, 256, 0, stream>>>(out, esum, emax, N);

  int attnBlocks = (N * 8 + 255) / 256;
  gat_node_attn<<<attnBlocks, 256, 0, stream>>>(ft, attn_l, attn_r, a1, a2, N);

  long long EH = E * 8;
  int edgeBlocks = (int)((EH + 255) / 256);
  gat_edge_max<<<edgeBlocks, 256, 0, stream>>>(ei, a1, a2, emax, E);
  gat_edge_expsum<<<edgeBlocks, 256, 0, stream>>>(ei, a1, a2, emax, esum, E);
  gat_edge_aggregate<<<edgeBlocks, 256, 0, stream>>>(ei, a1, a2, emax, esum, ft, out, E);
}